// CoopLatentTree_45681272160772
// MI455X (gfx1250) — compile-verified
//
#include <hip/hip_runtime.h>
#include <cstdint>

#define TL_B 64
#define TL_L 64
#define TL_D 512
#define TL_H 512

typedef float v2f __attribute__((ext_vector_type(2)));
typedef float v8f __attribute__((ext_vector_type(8)));
typedef int tl_v4i __attribute__((vector_size(16)));
typedef __attribute__((address_space(1))) tl_v4i* tl_gptr;
typedef __attribute__((address_space(3))) tl_v4i* tl_lptr;

#if __has_builtin(__builtin_amdgcn_global_load_async_to_lds_b128)
#define TL_ASYNC 1
#else
#define TL_ASYNC 0
#endif

__device__ __forceinline__ float tl_sigmoid(float x) {
  return 1.0f / (1.0f + expf(-x));
}

// one 16-byte global -> LDS copy (async path: GLOBAL_LOAD_ASYNC_TO_LDS_B128)
__device__ __forceinline__ void tl_cp_b128(const float* g, float* l) {
#if TL_ASYNC
  __builtin_amdgcn_global_load_async_to_lds_b128(
      (tl_gptr)(uintptr_t)g,
      (tl_lptr)(uint32_t)(uintptr_t)l, 0, 0);
#else
  *(float4*)l = *(const float4*)g;
#endif
}

__device__ __forceinline__ void tl_cp_wait() {
#if TL_ASYNC
#if __has_builtin(__builtin_amdgcn_s_wait_asynccnt)
  __builtin_amdgcn_s_wait_asynccnt(0);
#else
  asm volatile("s_wait_asynccnt 0" ::: "memory");
#endif
#endif
}

// ---------------------------------------------------------------------------
// Word projection: state = x @ w_word^T + b_word ; h = state[:,:H], c = rest
// grid (4096/64, 1024/128), block 256 (8 waves). fp32 WMMA 16x16x4.
// A tile row-major [64][20] (pad->16B chunks, conflict-free), B tile
// column-major [col][k] so each fragment is one aligned ds_load_b64.
// ---------------------------------------------------------------------------
__global__ __launch_bounds__(256) void tl_word_gemm(
    const float* __restrict__ x, const float* __restrict__ w_word,
    const float* __restrict__ b_word, float* __restrict__ h0,
    float* __restrict__ c0)
{
  __shared__ __align__(16) float at[64][20];
  __shared__ __align__(16) float btw[128][20];
  const int t = threadIdx.x;
  const int wid = t >> 5, lane = t & 31;
  const int waveM = wid & 3, waveN = wid >> 2;   // 4 x 2 wave grid
  const int kh = lane >> 4, nl = lane & 15;
  const int r0 = blockIdx.x * 64;
  const int nn0 = blockIdx.y * 128;

  // A loader: 64 rows x 4 chunks = 256 -> one b128 per thread per K-step
  const int arow = t >> 2;
  const int ach = (t & 3) * 4;
  const float* agp = x + (size_t)(r0 + arow) * TL_D + ach;
  float* alp = &at[arow][ach];

  // B loader: 128 cols x 4 chunks = 512 -> two b128 per thread per K-step
  const float* bgp[2];
  float* blp[2];
#pragma unroll
  for (int p = 0; p < 2; ++p) {
    const int idx = t + 256 * p;
    const int col = idx >> 2;
    const int ch = (idx & 3) * 4;
    bgp[p] = w_word + (size_t)(nn0 + col) * TL_D + ch;
    blp[p] = &btw[col][ch];
  }

  v8f acc[4] = {};
  for (int kk = 0; kk < TL_D; kk += 16) {
    __syncthreads();
    tl_cp_b128(agp + kk, alp);
    tl_cp_b128(bgp[0] + kk, blp[0]);
    tl_cp_b128(bgp[1] + kk, blp[1]);
    if (kk + 16 < TL_D) {
      __builtin_prefetch(agp + kk + 16, 0, 1);
      __builtin_prefetch(bgp[0] + kk + 16, 0, 1);
    }
    tl_cp_wait();
    __syncthreads();
#pragma unroll
    for (int s = 0; s < 4; ++s) {
      const int kb = 4 * s + 2 * kh;
      const v2f a = *(const v2f*)&at[waveM * 16 + nl][kb];
#pragma unroll
      for (int nt = 0; nt < 4; ++nt) {
        const v2f b = *(const v2f*)&btw[waveN * 64 + nt * 16 + nl][kb];
        acc[nt] = __builtin_amdgcn_wmma_f32_16x16x4_f32(
            false, a, false, b, (short)0, acc[nt], false, false);
      }
    }
  }
#pragma unroll
  for (int nt = 0; nt < 4; ++nt) {
    const int nn = nn0 + waveN * 64 + nt * 16 + nl;
    const float bias = b_word[nn];
#pragma unroll
    for (int v = 0; v < 8; ++v) {
      const int r = r0 + waveM * 16 + v + 8 * kh;   // C/D layout: M = v + 8*kh
      const float val = acc[nt][v] + bias;
      if (nn < TL_H) h0[(size_t)r * TL_H + nn] = val;
      else           c0[(size_t)r * TL_H + (nn - TL_H)] = val;
    }
  }
}

// ---------------------------------------------------------------------------
// Composition GEMM + fused TreeLSTM cell.
// A row (b,j) = [h_j | h_{j+1}] = contiguous 1024-float window of h_in.
// Each workgroup: 64 rows x 32 h-cols x all 5 gates (i,fl,fr,u,o), cell
// applied in the epilogue. grid (n, 16), block 256.
// ---------------------------------------------------------------------------
__global__ __launch_bounds__(256) void tl_comp_gemm_cell(
    const float* __restrict__ h_in, const float* __restrict__ c_in,
    const float* __restrict__ w_comp, const float* __restrict__ b_comp,
    float* __restrict__ cand_h, float* __restrict__ cand_c, int n)
{
  __shared__ __align__(16) float at[64][20];
  __shared__ __align__(16) float btc[160][20];   // column-major: [gate*32+hc][k]
  const int t = threadIdx.x;
  const int wid = t >> 5, lane = t & 31;
  const int waveM = wid & 3, waveN = wid >> 2;
  const int kh = lane >> 4, nl = lane & 15;
  const int r0 = blockIdx.x * 64;      // Mtot = 64*n, grid.x = n -> rows valid
  const int hblk = blockIdx.y * 32;

  // A loader: one b128 per thread per K-step
  const int arow = t >> 2;
  const int ach = (t & 3) * 4;
  const int rA = r0 + arow;
  const int bA = rA / n;
  const int jA = rA - bA * n;
  const float* agp = h_in + (size_t)(bA * TL_L + jA) * TL_H + ach;  // [h_j|h_j+1]
  float* alp = &at[arow][ach];

  // B loader: 160 cols x 4 chunks = 640 b128 over 3 guarded passes
  const float* bgp[3];
  float* blp[3];
  bool bok[3];
#pragma unroll
  for (int p = 0; p < 3; ++p) {
    const int idx = t + 256 * p;
    bok[p] = (idx < 640);
    const int ic = bok[p] ? idx : 639;
    const int col = ic >> 2;           // 0..159 = gate*32 + hc
    const int ch = (ic & 3) * 4;
    const int g = col >> 5;
    const int hc = col & 31;
    bgp[p] = w_comp + (size_t)(g * TL_H + hblk + hc) * (2 * TL_H) + ch;
    blp[p] = &btc[col][ch];
  }

  v8f acc[5] = {};
  for (int kk = 0; kk < 2 * TL_H; kk += 16) {
    __syncthreads();
    tl_cp_b128(agp + kk, alp);
#pragma unroll
    for (int p = 0; p < 3; ++p)
      if (bok[p]) tl_cp_b128(bgp[p] + kk, blp[p]);
    if (kk + 16 < 2 * TL_H) {
      __builtin_prefetch(agp + kk + 16, 0, 1);
      __builtin_prefetch(bgp[0] + kk + 16, 0, 1);
    }
    tl_cp_wait();
    __syncthreads();
#pragma unroll
    for (int s = 0; s < 4; ++s) {
      const int kb = 4 * s + 2 * kh;
      const v2f a = *(const v2f*)&at[waveM * 16 + nl][kb];
#pragma unroll
      for (int g = 0; g < 5; ++g) {
        const v2f b = *(const v2f*)&btc[g * 32 + waveN * 16 + nl][kb];
        acc[g] = __builtin_amdgcn_wmma_f32_16x16x4_f32(
            false, a, false, b, (short)0, acc[g], false, false);
      }
    }
  }

  // fused TreeLSTM cell epilogue
  const int hcol = hblk + waveN * 16 + nl;
  const float bi  = b_comp[0 * TL_H + hcol];
  const float bfl = b_comp[1 * TL_H + hcol];
  const float bfr = b_comp[2 * TL_H + hcol];
  const float bu  = b_comp[3 * TL_H + hcol];
  const float bo  = b_comp[4 * TL_H + hcol];
#pragma unroll
  for (int v = 0; v < 8; ++v) {
    const int r = r0 + waveM * 16 + v + 8 * kh;
    const int b = r / n;
    const int j = r - b * n;
    const float cl = c_in[(size_t)(b * TL_L + j) * TL_H + hcol];
    const float cr = c_in[(size_t)(b * TL_L + j + 1) * TL_H + hcol];
    const float gi  = acc[0][v] + bi;
    const float gfl = acc[1][v] + bfl;
    const float gfr = acc[2][v] + bfr;
    const float gu  = acc[3][v] + bu;
    const float go  = acc[4][v] + bo;
    const float cc = cl * tl_sigmoid(gfl + 1.0f) + cr * tl_sigmoid(gfr + 1.0f) +
                     tanhf(gu) * tl_sigmoid(gi);
    const float hh = tl_sigmoid(go) * tanhf(cc);
    cand_h[(size_t)r * TL_H + hcol] = hh;
    cand_c[(size_t)r * TL_H + hcol] = cc;
  }
}

// ---------------------------------------------------------------------------
// Candidate scores: one wave per row, shuffle reduction. Deterministic.
// ---------------------------------------------------------------------------
__global__ __launch_bounds__(256) void tl_score(
    const float* __restrict__ cand_h, const float* __restrict__ cq,
    float* __restrict__ scores)
{
  const int wid = threadIdx.x >> 5, lane = threadIdx.x & 31;
  const int r = blockIdx.x * 8 + wid;          // grid = 8n blocks, rows = 64n
  const float* row = cand_h + (size_t)r * TL_H;
  float s = 0.0f;
#pragma unroll
  for (int q = 0; q < TL_H / 32; ++q)
    s += row[lane + 32 * q] * cq[lane + 32 * q];
#pragma unroll
  for (int off = 16; off > 0; off >>= 1)
    s += __shfl_xor(s, off, 32);
  if (lane == 0) scores[r] = s * 0.04419417382415922f;  // 1/sqrt(512)
}

// ---------------------------------------------------------------------------
// Masked argmax per batch, first-index tie-break (matches jnp.argmax).
// sel[b] = chosen merge slot if "done", else INT_MAX (pure truncation).
// ---------------------------------------------------------------------------
__global__ void tl_argmax(const float* __restrict__ scores,
                          const int* __restrict__ length,
                          int* __restrict__ sel, int n, int it)
{
  __shared__ float sv[64];
  __shared__ int si[64];
  const int b = blockIdx.x, t = threadIdx.x;
  const int len = length[b];
  float v = -1000000000.0f;
  if (t < n && (it + 1 + t) < len) v = scores[b * n + t];
  sv[t] = v;
  si[t] = t;
  __syncthreads();
  for (int off = 32; off > 0; off >>= 1) {
    if (t < off) {
      const float v2 = sv[t + off];
      const int i2 = si[t + off];
      if (v2 > sv[t] || (v2 == sv[t] && i2 < si[t])) { sv[t] = v2; si[t] = i2; }
    }
    __syncthreads();
  }
  if (t == 0) sel[b] = ((it + 1) < len) ? si[0] : 0x7fffffff;
}

// ---------------------------------------------------------------------------
// Merge: h'[j] = j<k ? h[j] : (j==k ? cand[j] : h[j+1]).  float4 (b128) path.
// Ping-pong buffers remove the in-place shift hazard. grid = 32n, block 256.
// ---------------------------------------------------------------------------
__global__ __launch_bounds__(256) void tl_merge(
    const float4* __restrict__ h_in, const float4* __restrict__ c_in,
    const float4* __restrict__ cand_h, const float4* __restrict__ cand_c,
    const int* __restrict__ sel, float4* __restrict__ h_out,
    float4* __restrict__ c_out, int n)
{
  const int idx = blockIdx.x * 256 + threadIdx.x;   // 8192*n total
  const int col = idx & 127;                        // H/4 = 128
  const int rest = idx >> 7;                        // b*n + j
  const int b = rest / n;
  const int j = rest - b * n;
  const int k = sel[b];
  float4 hv, cv;
  if (j == k) {
    const size_t o = (size_t)rest * 128 + col;
    hv = cand_h[o];
    cv = cand_c[o];
  } else {
    const int jj = (j < k) ? j : j + 1;
    const size_t o = (size_t)(b * TL_L + jj) * 128 + col;
    hv = h_in[o];
    cv = c_in[o];
  }
  const size_t od = (size_t)(b * TL_L + j) * 128 + col;
  h_out[od] = hv;
  c_out[od] = cv;
}

// ---------------------------------------------------------------------------
// Output: concat(h[:,0,:], c[:,0,:]) flat.
// ---------------------------------------------------------------------------
__global__ void tl_finalize(const float* __restrict__ h_fin,
                            const float* __restrict__ c_fin,
                            float* __restrict__ out)
{
  const int idx = blockIdx.x * 256 + threadIdx.x;  // B*H = 32768
  const int b = idx >> 9;
  const int col = idx & 511;
  out[idx] = h_fin[(size_t)b * TL_L * TL_H + col];
  out[TL_B * TL_H + idx] = c_fin[(size_t)b * TL_L * TL_H + col];
}

// ---------------------------------------------------------------------------
extern "C" void kernel_launch(void* const* d_in, const int* in_sizes, int n_in,
                              void* d_out, int out_size, void* d_ws,
                              size_t ws_size, hipStream_t stream)
{
  (void)in_sizes; (void)n_in; (void)out_size; (void)ws_size;
  const float* x      = (const float*)d_in[0];
  const int*   length = (const int*)d_in[1];
  const float* w_word = (const float*)d_in[2];
  const float* b_word = (const float*)d_in[3];
  const float* w_comp = (const float*)d_in[4];
  const float* b_comp = (const float*)d_in[5];
  const float* cq     = (const float*)d_in[6];
  float* out = (float*)d_out;

  // workspace layout (~50 MB): h0 c0 h1 c1 cand_h cand_c scores sel
  const size_t SBUF  = (size_t)TL_B * TL_L * TL_H;
  const size_t SCAND = (size_t)TL_B * (TL_L - 1) * TL_H;
  float* h0 = (float*)d_ws;
  float* c0 = h0 + SBUF;
  float* h1 = c0 + SBUF;
  float* c1 = h1 + SBUF;
  float* cand_h = c1 + SBUF;
  float* cand_c = cand_h + SCAND;
  float* scores = cand_c + SCAND;
  int*   sel    = (int*)(scores + (size_t)TL_B * (TL_L - 1));

  tl_word_gemm<<<dim3((TL_B * TL_L) / 64, (2 * TL_H) / 128), 256, 0, stream>>>(
      x, w_word, b_word, h0, c0);

  float* hb[2] = {h0, h1};
  float* cb[2] = {c0, c1};
  for (int i = 0; i < TL_L - 1; ++i) {
    const int n = TL_L - 1 - i;
    const float* hi = hb[i & 1];
    const float* ci = cb[i & 1];
    float* ho = hb[(i + 1) & 1];
    float* co = cb[(i + 1) & 1];
    tl_comp_gemm_cell<<<dim3(n, TL_H / 32), 256, 0, stream>>>(
        hi, ci, w_comp, b_comp, cand_h, cand_c, n);
    tl_score<<<8 * n, 256, 0, stream>>>(cand_h, cq, scores);
    tl_argmax<<<TL_B, 64, 0, stream>>>(scores, length, sel, n, i);
    tl_merge<<<32 * n, 256, 0, stream>>>(
        (const float4*)hi, (const float4*)ci, (const float4*)cand_h,
        (const float4*)cand_c, sel, (float4*)ho, (float4*)co, n);
  }
  tl_finalize<<<(TL_B * TL_H) / 256, 256, 0, stream>>>(hb[1], cb[1], out);
}